// SpikingGustatory_48842368090339
// MI455X (gfx1250) — compile-verified
//
#include <hip/hip_runtime.h>
#include <stdint.h>

// ---------------------------------------------------------------------------
// SpikingGustatory forward: B=2,097,152 agents, 6 Izhikevich neurons x 10
// steps + 3x4 two-compartment predictive-coding column x 8 substeps.
// Memory-bound (~965 MB/launch vs ~3 GFLOP). CDNA5 path used: Tensor Data
// Mover bulk-stages each block's [10 x 256 x 6] f32 noise tile (60 KB) into
// LDS with a single tensor_load_to_lds, replacing 7680 strided VMEM loads.
// ---------------------------------------------------------------------------

#define B_AGENTS  2097152
#define NNEUR     6
#define NSTEPS    10
#define NCOLU     12
#define NSUB      8
#define BLK       256

typedef unsigned int u32x4 __attribute__((ext_vector_type(4)));
typedef int          i32x8 __attribute__((ext_vector_type(8)));
typedef int          i32x4 __attribute__((ext_vector_type(4)));

#if defined(__HIP_DEVICE_COMPILE__) && defined(__gfx1250__) && \
    __has_builtin(__builtin_amdgcn_tensor_load_to_lds) &&       \
    __has_builtin(__builtin_amdgcn_s_wait_tensorcnt)
#define USE_TDM 1
#else
#define USE_TDM 0
#endif

__global__ void __launch_bounds__(BLK)
SpikingGustatory_48842368090339_kernel(
    const unsigned char* __restrict__ eating,   // [B] bool
    const float* __restrict__ food_quality,     // [B]
    const float* __restrict__ food_distance,    // [B]
    const float* __restrict__ toxin_level,      // [B]
    const float* __restrict__ predicted_taste,  // [B]
    const float* __restrict__ noise,            // [STEPS, B, 6]
    const float* __restrict__ v0,               // [B, 6]
    const float* __restrict__ u0,               // [B, 6]
    const float* __restrict__ rate0,            // [B, 6]
    const float* __restrict__ vs0,              // [B, 12]
    const float* __restrict__ vd0,              // [B, 12]
    float* __restrict__ out)                    // [B, 9]
{
    const int tid = threadIdx.x;
    const int b0  = blockIdx.x * BLK;
    const int b   = b0 + tid;

#if USE_TDM
    // 60 KB LDS tile: [NSTEPS][BLK][NNEUR] f32, filled by the TDM.
    __shared__ float nzbuf[NSTEPS * BLK * NNEUR];

    if (tid < 32) {  // wave 0 only: TDM is a per-wave DMA, EXEC-independent
        const unsigned long long ga =
            (unsigned long long)(uintptr_t)noise +
            (unsigned long long)b0 * (NNEUR * 4);           // tile start (bytes)
        const unsigned ldsa =
            (unsigned)(unsigned long long)(uintptr_t)&nzbuf[0];  // LDS byte addr

        const unsigned          tensor_dim0 = (unsigned)B_AGENTS * NNEUR; // elems/row
        const unsigned          tensor_dim1 = NSTEPS;
        const unsigned          tile_dim0   = BLK * NNEUR;   // 1536 elems (6 KB)
        const unsigned          tile_dim1   = NSTEPS;        // 10 rows
        const unsigned long long stride0    = (unsigned long long)B_AGENTS * NNEUR;

        // ---- D# group 0 (CDNA5 ISA) ----
        u32x4 g0;
        g0[0] = 1u;                                          // count=1, user mode
        g0[1] = ldsa;                                        // lds_addr [63:32]
        g0[2] = (unsigned)(ga & 0xFFFFFFFFull);              // global_addr[31:0]
        g0[3] = ((unsigned)(ga >> 32) & 0x01FFFFFFu)         // global_addr[56:32]
              | (2u << 30);                                  // type=2 ("image")

        // ---- D# group 1 (§8.4): data_size=4B, 2-D tile ----
        i32x8 g1;
        g1[0] = (int)(2u << 16);                             // data_size=2 -> 4 bytes
        g1[1] = (int)((tensor_dim0 & 0xFFFFu) << 16);        // tensor_dim0[15:0]
        g1[2] = (int)((tensor_dim0 >> 16) |
                      ((tensor_dim1 & 0xFFFFu) << 16));      // td0[31:16] | td1[15:0]
        g1[3] = (int)((tensor_dim1 >> 16) |
                      (tile_dim0 << 16));                    // td1[31:16] | tile_dim0
        g1[4] = (int)(tile_dim1);                            // tile_dim1 | tile_dim2=0
        g1[5] = (int)(unsigned)(stride0 & 0xFFFFFFFFull);    // dim0_stride[31:0]
        g1[6] = (int)((unsigned)(stride0 >> 32) & 0xFFFFu);  // dim0_stride[47:32]
        g1[7] = 0;                                           // dim1_stride (unused, 2-D)

        i32x4 g2 = {0, 0, 0, 0};                             // dims 2/3 unused
        i32x4 g3 = {0, 0, 0, 0};
        i32x8 g4 = {0, 0, 0, 0, 0, 0, 0, 0};                 // 6-arg form: zero-filled

        __builtin_amdgcn_tensor_load_to_lds(g0, g1, g2, g3, g4, /*cpol=*/0);
    }
#endif

    // ---- Scalar inputs & taste model (overlaps the TDM transfer) ----
    const bool  eat  = eating[b] != 0;
    const float fq   = food_quality[b];
    const float fd   = food_distance[b];
    const float tox  = toxin_level[b];
    const float pred = predicted_taste[b];

    const float prox = fmaxf((30.0f - fd) * (1.0f / 30.0f), 0.0f);
    const bool  nearf = fd < 30.0f;

    const float amino  = eat ? fminf(1.0f, fq * 0.9f + 0.1f)
                             : (nearf ? fq * 0.3f * prox : 0.0f);
    const float umami  = eat ? fminf(1.0f, fq * 0.7f)
                             : (nearf ? fq * 0.2f * prox : 0.0f);
    const float bitter = eat ? fminf(1.0f, tox * 1.5f) : fminf(1.0f, tox * 0.5f);
    const float palat  = fminf(1.0f, fmaxf(-1.0f, amino * 0.5f + umami * 0.5f - bitter));
    const float spit   = (eat && (bitter > 0.4f) && (bitter > amino)) ? 1.0f : 0.0f;

    const float Iarr[NNEUR] = { amino * 15.0f, amino * 8.0f, bitter * 15.0f,
                                bitter * 8.0f, umami * 12.0f, umami * 8.0f };

    // ---- Load neuron state: [B,6] -> 8B-aligned float2 x3 each ----
    float v[NNEUR], u[NNEUR], r[NNEUR];
    {
        const float2* pv = reinterpret_cast<const float2*>(v0    + (size_t)b * NNEUR);
        const float2* pu = reinterpret_cast<const float2*>(u0    + (size_t)b * NNEUR);
        const float2* pr = reinterpret_cast<const float2*>(rate0 + (size_t)b * NNEUR);
#pragma unroll
        for (int k = 0; k < 3; ++k) {
            float2 a = pv[k]; v[2 * k] = a.x; v[2 * k + 1] = a.y;
            float2 c = pu[k]; u[2 * k] = c.x; u[2 * k + 1] = c.y;
            float2 d = pr[k]; r[2 * k] = d.x; r[2 * k + 1] = d.y;
        }
    }

    // ---- Load column state: [B,12] -> 16B-aligned float4 x3 each ----
    float vs[NCOLU], vd[NCOLU];
    {
        const float4* pvs = reinterpret_cast<const float4*>(vs0 + (size_t)b * NCOLU);
        const float4* pvd = reinterpret_cast<const float4*>(vd0 + (size_t)b * NCOLU);
#pragma unroll
        for (int k = 0; k < 3; ++k) {
            float4 a = pvs[k];
            vs[4 * k] = a.x; vs[4 * k + 1] = a.y; vs[4 * k + 2] = a.z; vs[4 * k + 3] = a.w;
            float4 c = pvd[k];
            vd[4 * k] = c.x; vd[4 * k + 1] = c.y; vd[4 * k + 2] = c.z; vd[4 * k + 3] = c.w;
        }
    }

#if USE_TDM
    if (tid < 32) __builtin_amdgcn_s_wait_tensorcnt(0);  // wave0: DMA complete
    __syncthreads();                                     // publish LDS to all waves
#endif

    // ---- 10 Izhikevich steps, noise from LDS tile (ds_load_b64 x3/step) ----
#pragma unroll
    for (int s = 0; s < NSTEPS; ++s) {
#if USE_TDM
        const float2* nr =
            reinterpret_cast<const float2*>(&nzbuf[(s * BLK + tid) * NNEUR]);
#else
        const float2* nr = reinterpret_cast<const float2*>(
            noise + ((size_t)s * B_AGENTS + b) * NNEUR);
#endif
        float nz[NNEUR];
#pragma unroll
        for (int k = 0; k < 3; ++k) {
            float2 a = nr[k]; nz[2 * k] = a.x; nz[2 * k + 1] = a.y;
        }
#pragma unroll
        for (int n = 0; n < NNEUR; ++n) {
            const float Iin = Iarr[n] + nz[n] * 0.3f - 1.0f;   // + I_TONIC
            const float vn  = v[n];
            const float v2  = vn + (0.04f * vn * vn + 5.0f * vn + 140.0f - u[n] + Iin);
            const float u2  = u[n] + 0.02f * (0.2f * vn - u[n]);
            const bool  spk = v2 >= 30.0f;
            v[n] = spk ? -65.0f : v2;
            u[n] = spk ? (u2 + 8.0f) : u2;
            r[n] = r[n] * 0.95f + (spk ? 0.05f : 0.0f);
        }
    }

    // ---- Two-compartment column, 8 substeps (dt = 1/8) ----
    const float s3[3] = { amino, bitter, umami };
    const float p3[3] = { pred * 0.5f, 0.0f, pred * 0.5f };
#pragma unroll
    for (int t = 0; t < NSUB; ++t) {
#pragma unroll
        for (int j = 0; j < NCOLU; ++j) {
            const int c = j >> 2;
            vd[j] += 0.125f * (p3[c] - vd[j]);
            vs[j] += 0.125f * (s3[c] + vd[j] - vs[j]);
        }
    }

    float pe_abs = 0.0f, prec_sum = 0.0f, fe = 0.0f;
#pragma unroll
    for (int c = 0; c < 3; ++c) {
        const float vsc = 0.25f * (vs[4 * c] + vs[4 * c + 1] + vs[4 * c + 2] + vs[4 * c + 3]);
        const float pe  = s3[c] - vsc;
        const float pe2 = pe * pe;
        const float pr  = 1.0f / (1.0f + pe2);
        pe_abs  += fabsf(pe);
        prec_sum += pr;
        fe      += 0.5f * (pr * pe2 + __logf(1.0f + pe2));  // -log(prec)=log(1+pe^2)
    }

    const float rate_mean =
        (r[0] + r[1] + r[2] + r[3] + r[4] + r[5]) * (1.0f / 6.0f);

    float* o = out + (size_t)b * 9;
    o[0] = amino;
    o[1] = bitter;
    o[2] = umami;
    o[3] = palat;
    o[4] = spit;
    o[5] = pe_abs * (1.0f / 3.0f);
    o[6] = prec_sum * (1.0f / 3.0f);
    o[7] = fe;
    o[8] = rate_mean;
}

extern "C" void kernel_launch(void* const* d_in, const int* in_sizes, int n_in,
                              void* d_out, int out_size, void* d_ws, size_t ws_size,
                              hipStream_t stream) {
    (void)in_sizes; (void)n_in; (void)out_size; (void)d_ws; (void)ws_size;
    const unsigned char* eating = (const unsigned char*)d_in[0];
    const float* food_quality    = (const float*)d_in[1];
    const float* food_distance   = (const float*)d_in[2];
    const float* toxin_level     = (const float*)d_in[3];
    const float* predicted_taste = (const float*)d_in[4];
    const float* noise           = (const float*)d_in[5];
    const float* v0              = (const float*)d_in[6];
    const float* u0              = (const float*)d_in[7];
    const float* rate0           = (const float*)d_in[8];
    const float* vs0             = (const float*)d_in[9];
    const float* vd0             = (const float*)d_in[10];
    float* out = (float*)d_out;

    dim3 grid(B_AGENTS / BLK), block(BLK);
    hipLaunchKernelGGL(SpikingGustatory_48842368090339_kernel, grid, block, 0, stream,
                       eating, food_quality, food_distance, toxin_level, predicted_taste,
                       noise, v0, u0, rate0, vs0, vd0, out);
}